// LlamaAttention_45749991637119
// MI455X (gfx1250) — compile-verified
//
#include <hip/hip_runtime.h>

// LlamaAttention (B=4, T=2048, HID=4096, H=64, D=64) — algebraically collapsed.
// Single-KV-head + per-token attention => scores constant over g => softmax
// uniform (exactly 1/64) => attn == v exactly.
// out = (hidden @ Wv.T) @ Wf.T, with Wf[i,d] = sum_h Wo[i, h*64+d].
// Bandwidth-bound (~270 MB streamed @ 23.3 TB/s); fp32 WMMA keeps full precision.

typedef __attribute__((ext_vector_type(2))) float v2f;
typedef __attribute__((ext_vector_type(4))) float v4f;
typedef __attribute__((ext_vector_type(8))) float v8f;
typedef __attribute__((ext_vector_type(4))) unsigned int u32x4;
typedef __attribute__((ext_vector_type(8))) unsigned int u32x8;
typedef __attribute__((address_space(3))) float lds_f;

#define HIDDEN 4096
#define HEADD  64
#define NTOK   8192      // B*T
#define KCHUNK 256
#define NCHUNK (HIDDEN / KCHUNK)   // 16
#define LDA    260       // padded LDS stride (260 % 64 == 4 -> conflict-free)
#define LDV    68        // padded LDS stride (68  % 64 == 4 -> conflict-free)

// CDNA5 async DMA: global -> LDS, no VGPR round-trip, tracked by ASYNCcnt.
__device__ __forceinline__ void async_cp_b128(lds_f* dst, const float* src) {
    asm volatile("global_load_async_to_lds_b128 %0, %1, off"
                 :: "v"(dst), "v"(src) : "memory");
}
__device__ __forceinline__ void wait_async0() {
    asm volatile("s_wait_asynccnt 0x0" ::: "memory");
}

__device__ __forceinline__ unsigned lds_byte_off(float* p) {
    return (unsigned)(unsigned long long)(lds_f*)p;   // addrspacecast -> 32-bit LDS offset
}

// Tensor Data Mover: one instruction DMAs a 16x256 f32 tile (row stride 4096)
// into LDS with 4-DWORD padding every 256 DWORDs (-> LDS pitch = LDA = 260).
// D# per CDNA5 ISA. Tracked by TENSORcnt.
__device__ __forceinline__ void tdm_load_tile16x256(unsigned lds_off, const float* gsrc) {
    unsigned long long ga = (unsigned long long)gsrc;
    u32x4 g0;
    g0[0] = 1u;                                   // count=1, user descriptor
    g0[1] = lds_off;                              // lds_addr (bytes)
    g0[2] = (unsigned)ga;                         // global_addr[31:0]
    g0[3] = (unsigned)((ga >> 32) & 0x01FFFFFFu)  // global_addr[56:32]
          | 0x80000000u;                          // type=2 ("image") in bits 127:126
    u32x8 g1;
    g1[0] = (2u << 16)                            // data_size = 4 bytes
          | (1u << 20)                            // pad_enable
          | (7u << 22)                            // pad_interval: every 256 DWORDs
          | (3u << 25);                           // pad_amount: 4 DWORDs
    g1[1] = (4096u & 0xFFFFu) << 16;              // tensor_dim0[15:0] @ bits 63:48
    g1[2] = (4096u >> 16) | (16u << 16);          // tensor_dim0[31:16] | tensor_dim1[15:0]
    g1[3] = (256u << 16);                         // tensor_dim1[31:16]=0 | tile_dim0=256
    g1[4] = 16u;                                  // tile_dim1=16 | tile_dim2=0
    g1[5] = 4096u;                                // tensor_dim0_stride[31:0]
    g1[6] = 0u;                                   // stride hi | dim1_stride lo
    g1[7] = 0u;
    asm volatile("tensor_load_to_lds %0, %1" :: "s"(g0), "s"(g1) : "memory");
}

// ---------------------------------------------------------------------------
// Kernel 1: fold Wo (4096x4096) over the 64 heads -> Wf (4096x64)
// ---------------------------------------------------------------------------
__global__ __launch_bounds__(256) void fold_wo(const float* __restrict__ Wo,
                                               float* __restrict__ Wf) {
    int t = blockIdx.x * 256 + threadIdx.x;   // 0 .. 4096*64-1
    int i = t >> 6;
    int d = t & 63;
    const float* p = Wo + (size_t)i * HIDDEN + d;
    float s = 0.f;
#pragma unroll
    for (int h = 0; h < 64; ++h) s += p[h * 64];
    Wf[t] = s;
}

// ---------------------------------------------------------------------------
// Kernel 2: v = hidden(8192x4096) @ Wv(64x4096)^T  -> v (8192x64)
// Block = 16 tokens, 4 waves; wave w owns output columns [16w, 16w+16).
// A-tiles double-buffered in LDS, filled by TDM (wave 0 issues; TENSORcnt
// in-order per wave => wait(1) after issuing chunk i+1 drains chunk i).
// ---------------------------------------------------------------------------
__global__ __launch_bounds__(128) void qv_gemm(const float* __restrict__ hidden,
                                               const float* __restrict__ Wv,
                                               float* __restrict__ v) {
    __shared__ float sA[2][16 * LDA];
    const int tid  = threadIdx.x;
    const int wave = tid >> 5;          // 0..3
    const int lane = tid & 31;
    const int m    = lane & 15;         // tile row / N column
    const int hi   = lane >> 4;         // half-wave select
    const int t0   = blockIdx.x * 16;   // token base
    const int d0   = wave * 16;         // output-column base

    // B[k][n] = Wv[n][k]; lane reads Wv row (d0+m), cols k+2*hi, k+2*hi+1
    const float* bptr = Wv + (size_t)(d0 + m) * HIDDEN + 2 * hi;
    const float* arow = hidden + (size_t)t0 * HIDDEN;

    const unsigned lds0 = lds_byte_off(&sA[0][0]);
    const unsigned lds1 = lds_byte_off(&sA[1][0]);

    if (wave == 0)                      // prologue: DMA chunk 0 into buffer 0
        tdm_load_tile16x256(lds0, arow);

    v8f acc = {};
    for (int i = 0; i < NCHUNK; ++i) {
        __syncthreads();                // all waves done reading buf[(i+1)&1]'s old data
        if (wave == 0) {
            if (i + 1 < NCHUNK) {       // overlap: DMA chunk i+1 behind compute of i
                tdm_load_tile16x256((i & 1) ? lds0 : lds1,
                                    arow + (size_t)(i + 1) * KCHUNK);
                __builtin_amdgcn_s_wait_tensorcnt(1);   // chunk i landed (in-order)
            } else {
                __builtin_amdgcn_s_wait_tensorcnt(0);
            }
        }
        __builtin_prefetch(bptr + (size_t)(i + 1) * KCHUNK, 0, 1);
        __syncthreads();                // publish buf[i&1]
        const float* sAbuf = &sA[i & 1][0];
        const float* bchunk = bptr + (size_t)i * KCHUNK;
#pragma unroll 4
        for (int kk = 0; kk < KCHUNK; kk += 4) {
            v2f a = *(const v2f*)(&sAbuf[m * LDA + kk + 2 * hi]);
            v2f b = *(const v2f*)(bchunk + kk);
            acc = __builtin_amdgcn_wmma_f32_16x16x4_f32(
                false, a, false, b, (short)0, acc, false, false);
        }
    }
    // C/D layout: VGPR r -> row (r + 8*hi), col = m
#pragma unroll
    for (int r = 0; r < 8; ++r)
        v[(size_t)(t0 + r + 8 * hi) * HEADD + d0 + m] = acc[r];
}

// ---------------------------------------------------------------------------
// Kernel 3: out = v(8192x64) @ Wf(4096x64)^T -> out (8192x4096)
// Block = 16 tokens x 128 cols, 8 waves (one 16x16 tile each), K = 64.
// v-tile staged with per-lane global_load_async_to_lds_b128 (ASYNCcnt path).
// ---------------------------------------------------------------------------
__global__ __launch_bounds__(256) void out_gemm(const float* __restrict__ v,
                                                const float* __restrict__ Wf,
                                                float* __restrict__ out) {
    __shared__ float sV[16 * LDV];
    const int tid  = threadIdx.x;
    const int wave = tid >> 5;                  // 0..7
    const int lane = tid & 31;
    const int m    = lane & 15;
    const int hi   = lane >> 4;
    const int t0   = blockIdx.x * 16;           // token base
    const int i0   = blockIdx.y * 128 + wave * 16;  // output-column base

    // Async v-tile fill: 16 x 64 floats = 256 x b128, one per thread
    {
        int r = tid >> 4;                       // 16 float4 per row
        int c = tid & 15;
        async_cp_b128((lds_f*)&sV[r * LDV + c * 4],
                      v + (size_t)(t0 + r) * HEADD + c * 4);
        wait_async0();
    }
    __syncthreads();

    // B[k][n] = Wf[n][k]; lane reads Wf row (i0+m), cols k+2*hi, +1 (contiguous)
    const float* bptr = Wf + (size_t)(i0 + m) * HEADD + 2 * hi;

    v8f acc = {};
#pragma unroll
    for (int k = 0; k < HEADD; k += 4) {
        v2f a = *(const v2f*)(&sV[m * LDV + k + 2 * hi]);
        v2f b = *(const v2f*)(bptr + k);
        acc = __builtin_amdgcn_wmma_f32_16x16x4_f32(
            false, a, false, b, (short)0, acc, false, false);
    }
#pragma unroll
    for (int r = 0; r < 8; ++r)
        out[(size_t)(t0 + r + 8 * hi) * HIDDEN + i0 + m] = acc[r];
}

// ---------------------------------------------------------------------------
extern "C" void kernel_launch(void* const* d_in, const int* in_sizes, int n_in,
                              void* d_out, int out_size, void* d_ws, size_t ws_size,
                              hipStream_t stream) {
    // setup_inputs order: hidden, cos, sin, Wq, Wk, Wv, Wo
    const float* hidden = (const float*)d_in[0];
    const float* Wv     = (const float*)d_in[5];
    const float* Wo     = (const float*)d_in[6];
    float* out = (float*)d_out;

    // Workspace: Wf (4096*64 = 1 MB) then v (8192*64 = 2 MB)
    float* Wf = (float*)d_ws;
    float* vv = Wf + (size_t)HIDDEN * HEADD;

    fold_wo<<<dim3((HIDDEN * HEADD) / 256), 256, 0, stream>>>(Wo, Wf);
    qv_gemm<<<dim3(NTOK / 16), 128, 0, stream>>>(hidden, Wv, vv);
    out_gemm<<<dim3(NTOK / 16, HIDDEN / 128), 256, 0, stream>>>(vv, Wf, out);
}